// LocalWindowAttention_73761768341561
// MI455X (gfx1250) — compile-verified
//
#include <hip/hip_runtime.h>

typedef __attribute__((ext_vector_type(2))) float v2f;
typedef __attribute__((ext_vector_type(8))) float v8f;
typedef __attribute__((ext_vector_type(2))) int   v2i;

#define THREADS   512     // 16 wave32; LDS limits to 1 block/WGP anyway
#define SO_STRIDE 258     // 256 + 2 pad -> 2-way LDS bank conflicts instead of 16-way
#define SQ_STRIDE 194     // 192 + 2 pad
#define SX_STRIDE 18      // 16 + 2 pad (also B-stage row stride), 8B aligned rows
#define SP_STRIDE 18
#define STAGE_FL  (144 * SX_STRIDE + 192 * SX_STRIDE)  // X[144][18] + B[192][18] = 6048 floats

// ---- CDNA5 async global->LDS staging (guarded; falls back to sync copy) ----
#if defined(__AMDGCN__) && __has_builtin(__builtin_amdgcn_global_load_async_to_lds_b64)
#define HAVE_ASYNC 1
#else
#define HAVE_ASYNC 0
#endif

__device__ __forceinline__ void ld64_to_lds(const float* g, float* l) {
#if HAVE_ASYNC
  __builtin_amdgcn_global_load_async_to_lds_b64(
      (__attribute__((address_space(1))) v2i*)g,
      (__attribute__((address_space(3))) v2i*)l, 0, 0);
#else
  *(v2f*)l = *(const v2f*)g;
#endif
}

__device__ __forceinline__ void stage_wait() {
#if HAVE_ASYNC
#if __has_builtin(__builtin_amdgcn_s_wait_asynccnt)
  __builtin_amdgcn_s_wait_asynccnt(0);
#else
  asm volatile("s_wait_asynccnt 0x0" ::: "memory");
#endif
#endif
}

// token m (0..143) of block blk -> flat token index (B=16,H=W=96,ws=3,32x32 windows)
__device__ __forceinline__ int tok_index(int blk, int m) {
  int win = m / 9;
  int t   = m - win * 9;
  int wg  = blk * 16 + win;
  int b   = wg >> 10;
  int rem = wg & 1023;
  int wh  = rem >> 5;
  int ww  = rem & 31;
  int tr  = t / 3;
  int row = wh * 3 + tr;
  int col = ww * 3 + (t - tr * 3);
  return (b * 96 + row) * 96 + col;
}

__device__ __forceinline__ v8f wmma4(v2f a, v2f b, v8f c) {
  // V_WMMA_F32_16X16X4_F32 : D = A(16x4,f32) * B(4x16,f32) + C(16x16,f32)
  return __builtin_amdgcn_wmma_f32_16x16x4_f32(false, a, false, b, (short)0, c, false, false);
}

__global__ __launch_bounds__(THREADS) void win_attn_fused(
    const float* __restrict__ x,      // [147456, 256]
    const float* __restrict__ Wqkv,   // [768, 256]
    const float* __restrict__ Wproj,  // [256, 256]
    float* __restrict__ out)          // [147456, 256]
{
  __shared__ float sO[144 * SO_STRIDE];   // concat attention output (GEMM2 A)
  __shared__ float sQKV[152 * SQ_STRIDE]; // per-head q|k|v (+8 zero pad rows)
  __shared__ float sStage[2 * STAGE_FL];  // double-buffered K-chunk staging; P-bounce aliases buf0

  const int tid  = threadIdx.x;
  const int blk  = blockIdx.x;
  const int wave = tid >> 5;   // 0..15
  const int lane = tid & 31;
  const int half = lane >> 4;
  const int l16  = lane & 15;

  // warm L2 for the (re-read) weights
  __builtin_prefetch(Wproj + (size_t)tid * 128, 0, 1);   // 512 * 512B = 256 KB
  __builtin_prefetch(Wqkv  + (size_t)tid * 384, 0, 1);

  // zero sQKV pad rows 144..151 (last window's 16-row tiles read up to row 150)
  for (int e = tid; e < 8 * SQ_STRIDE; e += THREADS)
    sQKV[144 * SQ_STRIDE + e] = 0.0f;

  // X staging map: b64 pair index e = tid + i*512 -> row m = e>>3, pair p = tid&7
  int gX[3];
  {
    int p2 = (tid & 7) * 2;
    #pragma unroll
    for (int i = 0; i < 3; ++i) {
      int m = (tid >> 3) + i * 64;                  // 0..191, valid < 144
      gX[i] = (m < 144) ? tok_index(blk, m) * 256 + p2 : 0;
    }
  }
  __syncthreads();

  const float scale = 0.125f;                       // 64^-0.5

  for (int h = 0; h < 4; ++h) {
    // head-h B row map: col n (0..191) -> Wqkv row (n/64)*256 + h*64 + (n%64)
    int gB[3];
    {
      int p2 = (tid & 7) * 2;
      #pragma unroll
      for (int i = 0; i < 3; ++i) {
        int n  = (tid >> 3) + i * 64;               // 0..191
        int wr = (n >> 6) * 256 + h * 64 + (n & 63);
        gB[i] = wr * 256 + p2;
      }
    }

    v8f acc[7];
    {
      v8f zz = {};
      #pragma unroll
      for (int i = 0; i < 7; ++i) acc[i] = zz;
    }

    // ---- GEMM1: qkv_h[144x192] = X[144x256] @ Wqkv_h^T, K in 16 chunks of 16 ----
    // issue chunk 0
    {
      float* bX = sStage;
      float* bB = bX + 144 * SX_STRIDE;
      int p2 = (tid & 7) * 2;
      #pragma unroll
      for (int i = 0; i < 3; ++i) {
        int m = (tid >> 3) + i * 64;
        if (m < 144) ld64_to_lds(x + gX[i], bX + m * SX_STRIDE + p2);
      }
      #pragma unroll
      for (int i = 0; i < 3; ++i) {
        int n = (tid >> 3) + i * 64;
        ld64_to_lds(Wqkv + gB[i], bB + n * SX_STRIDE + p2);
      }
    }
    for (int kc = 0; kc < 16; ++kc) {
      stage_wait();
      __syncthreads();
      if (kc + 1 < 16) {                            // issue next chunk into other buffer
        float* bX = &sStage[((kc + 1) & 1) * STAGE_FL];
        float* bB = bX + 144 * SX_STRIDE;
        int koff = (kc + 1) * 16;
        int p2 = (tid & 7) * 2;
        #pragma unroll
        for (int i = 0; i < 3; ++i) {
          int m = (tid >> 3) + i * 64;
          if (m < 144) ld64_to_lds(x + gX[i] + koff, bX + m * SX_STRIDE + p2);
        }
        #pragma unroll
        for (int i = 0; i < 3; ++i) {
          int n = (tid >> 3) + i * 64;
          ld64_to_lds(Wqkv + gB[i] + koff, bB + n * SX_STRIDE + p2);
        }
      }
      // compute current chunk: 9 Mtiles x 12 Ntiles = 108 tiles over 16 waves
      const float* bX = &sStage[(kc & 1) * STAGE_FL];
      const float* bB = bX + 144 * SX_STRIDE;
      #pragma unroll
      for (int lt = 0; lt < 7; ++lt) {
        int t = wave + lt * 16;
        if (t < 108) {
          int mt = t / 12, nt = t - (t / 12) * 12;
          const float* ax = &bX[(mt * 16 + l16) * SX_STRIDE + 2 * half];
          const float* bb = &bB[(nt * 16 + l16) * SX_STRIDE + 2 * half];
          #pragma unroll
          for (int ks = 0; ks < 4; ++ks) {
            v2f a = *(const v2f*)(ax + ks * 4);     // A: row=l16, k = 4ks+2h+{0,1}
            v2f b = *(const v2f*)(bb + ks * 4);     // B: col=l16, k = 4ks+2h+{0,1}
            acc[lt] = wmma4(a, b, acc[lt]);
          }
        }
      }
    }
    // spill qkv_h accumulators (C-layout: row = r + 8*half, col = l16)
    #pragma unroll
    for (int lt = 0; lt < 7; ++lt) {
      int t = wave + lt * 16;
      if (t < 108) {
        int mt = t / 12, nt = t - (t / 12) * 12;
        #pragma unroll
        for (int r = 0; r < 8; ++r)
          sQKV[(mt * 16 + r + 8 * half) * SQ_STRIDE + nt * 16 + l16] = acc[lt][r];
      }
    }
    __syncthreads();

    // ---- attention: one 9-token window per wave (scores padded to 16x16) ----
    {
      int rb0 = wave * 9;
      v8f s = {};
      const float* qrow = &sQKV[(rb0 + l16) * SQ_STRIDE + 2 * half];
      #pragma unroll
      for (int ks = 0; ks < 16; ++ks) {             // K = head_dim = 64
        v2f a = *(const v2f*)(qrow + ks * 4);       // q  : A row i = l16
        v2f b = *(const v2f*)(qrow + 64 + ks * 4);  // k^T: B col j = l16
        s = wmma4(a, b, s);
      }
      // softmax over j = l16 (mask j>=9), 16-lane shuffle reductions
      float p[8];
      #pragma unroll
      for (int r = 0; r < 8; ++r) {
        float v  = s[r] * scale;
        float vm = (l16 < 9) ? v : -1e30f;
        float mx = vm;
        mx = fmaxf(mx, __shfl_xor(mx, 1, 16));
        mx = fmaxf(mx, __shfl_xor(mx, 2, 16));
        mx = fmaxf(mx, __shfl_xor(mx, 4, 16));
        mx = fmaxf(mx, __shfl_xor(mx, 8, 16));
        float e = (l16 < 9) ? __expf(vm - mx) : 0.0f;
        float sum = e;
        sum += __shfl_xor(sum, 1, 16);
        sum += __shfl_xor(sum, 2, 16);
        sum += __shfl_xor(sum, 4, 16);
        sum += __shfl_xor(sum, 8, 16);
        p[r] = e / sum;
      }
      // bounce P through LDS (C-layout -> A-layout); alias idle staging buffer 0
      float* pw = &sStage[wave * 16 * SP_STRIDE];
      #pragma unroll
      for (int r = 0; r < 8; ++r)
        pw[(r + 8 * half) * SP_STRIDE + l16] = p[r];
      // O = P @ V  (K = 16 padded tokens; masked P cols are exactly 0)
      #pragma unroll
      for (int nt = 0; nt < 4; ++nt) {
        v8f o = {};
        #pragma unroll
        for (int ks = 0; ks < 4; ++ks) {
          v2f a = *(const v2f*)&pw[l16 * SP_STRIDE + ks * 4 + 2 * half];
          int j0 = ks * 4 + 2 * half;
          v2f b;
          b.x = sQKV[(rb0 + j0) * SQ_STRIDE + 128 + nt * 16 + l16];
          b.y = sQKV[(rb0 + j0 + 1) * SQ_STRIDE + 128 + nt * 16 + l16];
          o = wmma4(a, b, o);
        }
        #pragma unroll
        for (int r = 0; r < 8; ++r) {
          int i = r + 8 * half;
          if (i < 9)
            sO[(rb0 + i) * SO_STRIDE + h * 64 + nt * 16 + l16] = o[r];
        }
      }
    }
    __syncthreads();
  } // heads

  // ---- GEMM2: out[144x256] = sO @ Wproj^T, K in 16 chunks of 16 ----
  v8f acc2[9];
  {
    v8f zz = {};
    #pragma unroll
    for (int i = 0; i < 9; ++i) acc2[i] = zz;
  }
  int gP[4];
  {
    int p2 = (tid & 7) * 2;
    #pragma unroll
    for (int i = 0; i < 4; ++i) {
      int c = (tid >> 3) + i * 64;                  // 0..255
      gP[i] = c * 256 + p2;                         // Wproj[c][K + p2..]
    }
  }
  // issue chunk 0
  {
    float* bB2 = sStage;
    int p2 = (tid & 7) * 2;
    #pragma unroll
    for (int i = 0; i < 4; ++i) {
      int c = (tid >> 3) + i * 64;
      ld64_to_lds(Wproj + gP[i], bB2 + c * SX_STRIDE + p2);
    }
  }
  for (int kc = 0; kc < 16; ++kc) {
    stage_wait();
    __syncthreads();
    if (kc + 1 < 16) {
      float* bB2 = &sStage[((kc + 1) & 1) * STAGE_FL];
      int koff = (kc + 1) * 16;
      int p2 = (tid & 7) * 2;
      #pragma unroll
      for (int i = 0; i < 4; ++i) {
        int c = (tid >> 3) + i * 64;
        ld64_to_lds(Wproj + gP[i] + koff, bB2 + c * SX_STRIDE + p2);
      }
    }
    // 9 Mtiles x 16 Ntiles = 144 tiles, 9 per wave
    const float* bB2 = &sStage[(kc & 1) * STAGE_FL];
    #pragma unroll
    for (int lt = 0; lt < 9; ++lt) {
      int t  = wave + lt * 16;
      int mt = t >> 4, nt = t & 15;
      const float* ao = &sO[(mt * 16 + l16) * SO_STRIDE + kc * 16 + 2 * half];
      const float* bb = &bB2[(nt * 16 + l16) * SX_STRIDE + 2 * half];
      #pragma unroll
      for (int ks = 0; ks < 4; ++ks) {
        v2f a = *(const v2f*)(ao + ks * 4);
        v2f b = *(const v2f*)(bb + ks * 4);
        acc2[lt] = wmma4(a, b, acc2[lt]);
      }
    }
  }

  // scatter back through the window map
  #pragma unroll
  for (int lt = 0; lt < 9; ++lt) {
    int t  = wave + lt * 16;
    int mt = t >> 4, nt = t & 15;
    #pragma unroll
    for (int r = 0; r < 8; ++r) {
      int m = mt * 16 + r + 8 * half;               // < 144
      int g = tok_index(blk, m);
      out[g * 256 + nt * 16 + l16] = acc2[lt][r];
    }
  }
}

extern "C" void kernel_launch(void* const* d_in, const int* in_sizes, int n_in,
                              void* d_out, int out_size, void* d_ws, size_t ws_size,
                              hipStream_t stream) {
  (void)in_sizes; (void)n_in; (void)out_size; (void)d_ws; (void)ws_size;
  const float* x     = (const float*)d_in[0];
  const float* Wqkv  = (const float*)d_in[1];
  const float* Wproj = (const float*)d_in[2];
  float* out = (float*)d_out;
  // 16384 windows / 16 per block = 1024 blocks, 512 threads (16 wave32)
  win_attn_fused<<<dim3(1024), dim3(THREADS), 0, stream>>>(x, Wqkv, Wproj, out);
}